// PathGuidedAggregator_3770981286091
// MI455X (gfx1250) — compile-verified
//
#include <hip/hip_runtime.h>
#include <hip/hip_bf16.h>

#define N_ENT   100000
#define DIM     256
#define P_PATHS 5
#define E_EDGES 800000
#define HID     128   // DIM/2

typedef __attribute__((ext_vector_type(2))) float v2f;
typedef __attribute__((ext_vector_type(8))) float v8f;

// ---------------------------------------------------------------------------
// 1) Per-path row degree: cnt[p*N + r] += 1  (f32 atomic, zeroed beforehand)
// ---------------------------------------------------------------------------
__global__ void pga_count_kernel(const int* __restrict__ rows,
                                 float* __restrict__ cnt) {
    int i = blockIdx.x * blockDim.x + threadIdx.x;     // P*E = 4,000,000
    if (i >= P_PATHS * E_EDGES) return;
    int p = i / E_EDGES;
    int r = rows[i];
    atomicAdd(&cnt[p * N_ENT + r], 1.0f);
}

// ---------------------------------------------------------------------------
// 2) inv = 1 / max(cnt, 1)   (in place)
// ---------------------------------------------------------------------------
__global__ void pga_inv_kernel(float* __restrict__ cnt) {
    int i = blockIdx.x * blockDim.x + threadIdx.x;
    if (i >= P_PATHS * N_ENT) return;
    cnt[i] = 1.0f / fmaxf(cnt[i], 1.0f);
}

// ---------------------------------------------------------------------------
// 3) SpMM scatter: pf[(row*P + p)*D + d] += embeds[col*D + d] * inv[p,row]
//    Block = 256 threads = 4 edges x 64 lanes, each lane moves a float4.
//    Gather is fully coalesced (1 KB per edge); atomics land on consecutive
//    DWORDs of the destination row.
// ---------------------------------------------------------------------------
__global__ void pga_scatter_kernel(const int* __restrict__ rows,
                                   const int* __restrict__ cols,
                                   const float* __restrict__ embeds,
                                   const float* __restrict__ inv,
                                   float* __restrict__ pf) {
    int sub  = threadIdx.x >> 6;          // edge within block (0..3)
    int lane = threadIdx.x & 63;          // float4 chunk (0..63)
    long long eg = (long long)blockIdx.x * 4 + sub;
    if (eg >= (long long)P_PATHS * E_EDGES) return;
    int p = (int)(eg / E_EDGES);
    int r = rows[eg];
    int c = cols[eg];
    float w = inv[p * N_ENT + r];
    const float4* src = (const float4*)(embeds + (long long)c * DIM);
    float4 v = src[lane];
    float* dst = pf + ((long long)r * P_PATHS + p) * DIM + lane * 4;
    atomicAdd(dst + 0, v.x * w);
    atomicAdd(dst + 1, v.y * w);
    atomicAdd(dst + 2, v.z * w);
    atomicAdd(dst + 3, v.w * w);
}

// ---------------------------------------------------------------------------
// 4) Attention scores via fp32 WMMA:
//    S = tanh(pf @ W1 + b1)  [M=500000 x 128], score = S @ W2 + b2
//    One wave per 16-row M-tile; 8 v8f accumulators cover all 128 cols.
//    K-loop: 64 x V_WMMA_F32_16X16X4_F32 per accumulator set.
// ---------------------------------------------------------------------------
__global__ void pga_score_kernel(const float* __restrict__ pf,
                                 const float* __restrict__ W1,
                                 const float* __restrict__ b1,
                                 const float* __restrict__ W2,
                                 const float* __restrict__ b2,
                                 float* __restrict__ scores) {
    const int lane = threadIdx.x & 31;
    const int wave = threadIdx.x >> 5;
    const int tile = blockIdx.x * (blockDim.x >> 5) + wave;   // 16-row tile
    const int M_TILES = (N_ENT * P_PATHS) / 16;               // 31250 exactly
    if (tile >= M_TILES) return;   // wave-uniform -> EXEC stays all-1s

    const int half = lane >> 4;    // 0: lanes 0-15, 1: lanes 16-31
    const int l16  = lane & 15;

    // A fragment source: row (tile*16 + l16), K offset half*2 per §7.12.2
    const float* Arow = pf + ((long long)tile * 16 + l16) * DIM + half * 2;

    v8f acc[8];
#pragma unroll
    for (int t = 0; t < 8; ++t)
        acc[t] = (v8f){0.f, 0.f, 0.f, 0.f, 0.f, 0.f, 0.f, 0.f};

    for (int k = 0; k < DIM; k += 4) {
        v2f a;
        a.x = Arow[k + 0];          // K = k   (+2 for upper half-wave)
        a.y = Arow[k + 1];          // K = k+1 (+2 for upper half-wave)
#pragma unroll
        for (int t = 0; t < 8; ++t) {
            int h = t * 16 + l16;   // output column (N)
            v2f b;
            b.x = W1[(k + half * 2 + 0) * HID + h];
            b.y = W1[(k + half * 2 + 1) * HID + h];
            acc[t] = __builtin_amdgcn_wmma_f32_16x16x4_f32(
                false, a, false, b, (short)0, acc[t], false, false);
        }
    }

    // acc[t][v] at this lane = S[m][n], m = tile*16 + v + half*8, n = t*16+l16
    float partial[8];
#pragma unroll
    for (int v = 0; v < 8; ++v) partial[v] = 0.0f;
#pragma unroll
    for (int t = 0; t < 8; ++t) {
        int h = t * 16 + l16;
        float bb = b1[h];
        float w2 = W2[h];
#pragma unroll
        for (int v = 0; v < 8; ++v)
            partial[v] += tanhf(acc[t][v] + bb) * w2;
    }
    // Reduce over the 16 lanes of each half-wave (column dimension of S)
#pragma unroll
    for (int v = 0; v < 8; ++v) {
        float s = partial[v];
        s += __shfl_xor(s, 8, 16);
        s += __shfl_xor(s, 4, 16);
        s += __shfl_xor(s, 2, 16);
        s += __shfl_xor(s, 1, 16);
        partial[v] = s;
    }
    if (l16 == 0) {
        float bb2 = b2[0];
#pragma unroll
        for (int v = 0; v < 8; ++v)
            scores[tile * 16 + v + half * 8] = partial[v] + bb2;
    }
}

// ---------------------------------------------------------------------------
// 5) Softmax over P=5 paths + weighted sum -> out[N, D]
//    One block per entity, thread d handles dimension d.
// ---------------------------------------------------------------------------
__global__ void pga_combine_kernel(const float* __restrict__ pf,
                                   const float* __restrict__ scores,
                                   float* __restrict__ out) {
    int n = blockIdx.x;
    int d = threadIdx.x;              // DIM = 256 threads
    float s[P_PATHS];
    float mx = -3.4e38f;
#pragma unroll
    for (int p = 0; p < P_PATHS; ++p) {
        s[p] = scores[n * P_PATHS + p];
        mx = fmaxf(mx, s[p]);
    }
    float denom = 0.0f;
#pragma unroll
    for (int p = 0; p < P_PATHS; ++p) {
        s[p] = __expf(s[p] - mx);
        denom += s[p];
    }
    float invd = 1.0f / denom;
    float acc = 0.0f;
#pragma unroll
    for (int p = 0; p < P_PATHS; ++p)
        acc += s[p] * pf[((long long)n * P_PATHS + p) * DIM + d];
    out[(long long)n * DIM + d] = acc * invd;
}

// ---------------------------------------------------------------------------
extern "C" void kernel_launch(void* const* d_in, const int* in_sizes, int n_in,
                              void* d_out, int out_size, void* d_ws, size_t ws_size,
                              hipStream_t stream) {
    const float* embeds = (const float*)d_in[0];   // [N, D]
    const int*   rows   = (const int*)  d_in[1];   // [P, E]
    const int*   cols   = (const int*)  d_in[2];   // [P, E]
    const float* W1     = (const float*)d_in[3];   // [D, H]
    const float* b1     = (const float*)d_in[4];   // [H]
    const float* W2     = (const float*)d_in[5];   // [H, 1]
    const float* b2     = (const float*)d_in[6];   // [1]
    float* out = (float*)d_out;                    // [N, D]

    char* ws = (char*)d_ws;
    size_t pf_bytes  = (size_t)N_ENT * P_PATHS * DIM * sizeof(float);   // 512 MB
    size_t cnt_bytes = (size_t)P_PATHS * N_ENT * sizeof(float);         // 2 MB
    float* pf     = (float*)ws;
    float* cnt    = (float*)(ws + pf_bytes);
    float* scores = (float*)(ws + pf_bytes + cnt_bytes);                // 2 MB

    hipMemsetAsync(pf,  0, pf_bytes,  stream);
    hipMemsetAsync(cnt, 0, cnt_bytes, stream);

    {
        int total = P_PATHS * E_EDGES;
        pga_count_kernel<<<(total + 255) / 256, 256, 0, stream>>>(rows, cnt);
    }
    {
        int total = P_PATHS * N_ENT;
        pga_inv_kernel<<<(total + 255) / 256, 256, 0, stream>>>(cnt);
    }
    pga_scatter_kernel<<<(P_PATHS * E_EDGES) / 4, 256, 0, stream>>>(
        rows, cols, embeds, cnt, pf);
    {
        int mtiles = (N_ENT * P_PATHS) / 16;        // 31250
        int blocks = (mtiles + 7) / 8;              // 8 waves per block
        pga_score_kernel<<<blocks, 256, 0, stream>>>(pf, W1, b1, W2, b2, scores);
    }
    pga_combine_kernel<<<N_ENT, 256, 0, stream>>>(pf, scores, out);
}